// RippleDecoder_18528488915298
// MI455X (gfx1250) — compile-verified
//
#include <hip/hip_runtime.h>
#include <hip/hip_bf16.h>

// ---------------------------------------------------------------------------
// RippleDecoder for MI455X (gfx1250, wave32, WMMA + TDM)
//   Phase 1: 4-row MLP via V_WMMA_F32_16X16X4_F32 (f32-exact tensor path)
//   Phase 2: ripple decode, one wave per sequence position; per-batch params
//            staged to LDS with TENSOR_LOAD_TO_LDS (Tensor Data Mover)
// ---------------------------------------------------------------------------

#define B_ROWS    4
#define IN_DIM    512
#define HID       1024
#define SEQ       4096
#define RH        64
#define RL        2
#define MID_DIM   (2 * RH * RH + (RH + 1) * RH)      // 12352
#define OUT_DIM   (2 * RH + (RH + 1))                // 193
#define MLP_OUT_C (RL * MID_DIM + OUT_DIM)           // 24897

typedef float v2f __attribute__((ext_vector_type(2)));
typedef float v8f __attribute__((ext_vector_type(8)));
typedef unsigned int v4u __attribute__((ext_vector_type(4)));
typedef int v8i __attribute__((ext_vector_type(8)));
typedef int v4i __attribute__((ext_vector_type(4)));

__device__ __forceinline__ float gelu_f(float x) {
    // exact GELU: x * 0.5 * (1 + erf(x / sqrt(2)))
    return 0.5f * x * (1.0f + erff(x * 0.70710678118654752440f));
}

__device__ __forceinline__ float wave_sum32(float v) {
    #pragma unroll
    for (int m = 16; m >= 1; m >>= 1) v += __shfl_xor(v, m, 32);
    return v;
}

// ---------------------------------------------------------------------------
// Tensor Data Mover: DMA one contiguous row of `nelem` f32 from global
// memory into LDS at byte offset `lds_off`. D# built per CDNA5 ISA §8:
//   group0: count=1 | lds_addr | global_addr[56:0] | type=2 ("image")
//   group1: data_size=4B, tensor_dim0=nelem, tensor_dim1=1,
//           tile_dim0=nelem (<=65535), tile_dim1=1, dim0_stride=nelem
// Groups 2/3 zero (2D descriptor). Tracked by TENSORcnt.
// 6-arg builtin form (clang-23 / therock-10.0 headers).
// ---------------------------------------------------------------------------
__device__ __forceinline__ void tdm_load_row_to_lds(const float* gsrc,
                                                    unsigned lds_off,
                                                    unsigned nelem) {
    const unsigned long long ga = (unsigned long long)(uintptr_t)gsrc;
    v4u g0;
    g0[0] = 1u;                                          // count=1 (valid D#)
    g0[1] = lds_off;                                     // lds_addr (bytes)
    g0[2] = (unsigned)(ga & 0xffffffffu);                // global_addr[31:0]
    g0[3] = (unsigned)((ga >> 32) & 0x01ffffffu)         // global_addr[56:32]
          | (2u << 30);                                  // type=2
    v8i g1;
    g1[0] = (2 << 16);                                   // data_size=2 -> 4B
    g1[1] = (int)((nelem & 0xffffu) << 16);              // tensor_dim0[15:0]
    g1[2] = (int)((nelem >> 16) & 0xffffu)               // tensor_dim0[31:16]
          | (1 << 16);                                   // tensor_dim1 = 1
    g1[3] = (int)((nelem & 0xffffu) << 16);              // tile_dim0
    g1[4] = 1;                                           // tile_dim1 = 1
    g1[5] = (int)nelem;                                  // tensor_dim0_stride
    g1[6] = 0;
    g1[7] = 0;
    const v4i z4 = {0, 0, 0, 0};
    const v8i z8 = {0, 0, 0, 0, 0, 0, 0, 0};
    __builtin_amdgcn_tensor_load_to_lds(g0, g1, z4, z4, z8, 0);
}

// ---------------------------------------------------------------------------
// GEMM: out[4,N] = act(A[4,K] @ W[K,N] + bias[N])  via WMMA f32 16x16x4.
// One wave per 16-column tile. M=16 with rows 4..15 zero-padded.
// All load addresses are clamped in-range so masking lowers to v_cndmask
// (no EXEC-divergent conditional-load blocks in the inner loop).
// ---------------------------------------------------------------------------
__global__ __launch_bounds__(128)
void gemm4_wmma(const float* __restrict__ A, const float* __restrict__ W,
                const float* __restrict__ bias, float* __restrict__ out,
                int K, int N, int act)
{
    const int wave = (blockIdx.x * blockDim.x + threadIdx.x) >> 5;
    const int lane = threadIdx.x & 31;
    const int n0 = wave * 16;
    if (n0 >= N) return;                       // wave-uniform exit

    const int kh   = (lane < 16) ? 0 : 2;      // K sub-offset for this half-wave
    const int m    = lane & 15;                // A row held by this lane
    const int col  = n0 + (lane & 15);         // B/C column held by this lane
    const bool mOK   = (m < B_ROWS);
    const bool colOK = (col < N);
    const int  mc    = mOK   ? m   : 0;        // clamped (always-valid) row
    const int  colc  = colOK ? col : (N - 1);  // clamped (always-valid) col

    const float* __restrict__ Arow = A + (size_t)mc * K;   // 8B-aligned rows
    const float* __restrict__ Wcol = W + colc;

    v8f c = {};
    #pragma unroll 2
    for (int k0 = 0; k0 < K; k0 += 4) {
        const int ka = k0 + kh;
        // A fragment: contiguous pair -> one global_load_b64, then cndmask
        const v2f av = *(const v2f*)(Arow + ka);
        v2f a, b;
        a.x = mOK ? av.x : 0.0f;
        a.y = mOK ? av.y : 0.0f;
        // B fragment: two K-rows of the same column (stride N)
        const float bx = Wcol[(size_t)ka * N];
        const float by = Wcol[(size_t)(ka + 1) * N];
        b.x = colOK ? bx : 0.0f;
        b.y = colOK ? by : 0.0f;
        if ((k0 & 63) == 0 && (k0 + 68) < K)   // wave-uniform scalar branch
            __builtin_prefetch(Wcol + (size_t)(ka + 64) * N, 0, 1);
        // D = A*B + C, f32 exact (v_wmma_f32_16x16x4_f32)
        c = __builtin_amdgcn_wmma_f32_16x16x4_f32(
                false, a, false, b, (short)0, c, false, false);
    }

    // C/D layout: VGPR r holds M=r (lanes 0-15) and M=r+8 (lanes 16-31).
    if (lane < 16 && colOK) {
        #pragma unroll
        for (int r = 0; r < B_ROWS; ++r) {
            float v = c[r] + bias[col];
            if (act == 1) v = gelu_f(v);
            out[(size_t)r * N + col] = v;
        }
    }
}

// ---------------------------------------------------------------------------
// LayerNorm (ddof=0, eps=1e-5) + GELU over rows of length N. One block/row.
// ---------------------------------------------------------------------------
__global__ __launch_bounds__(256)
void ln_gelu_kernel(const float* __restrict__ in, float* __restrict__ out,
                    const float* __restrict__ g, const float* __restrict__ bt, int N)
{
    const int row = blockIdx.x;
    const float* r = in + (size_t)row * N;
    float s = 0.0f, s2 = 0.0f;
    for (int i = threadIdx.x; i < N; i += blockDim.x) {
        float v = r[i]; s += v; s2 += v * v;
    }
    s  = wave_sum32(s);
    s2 = wave_sum32(s2);
    __shared__ float sh0[8], sh1[8];
    const int wid = threadIdx.x >> 5, lane = threadIdx.x & 31;
    if (lane == 0) { sh0[wid] = s; sh1[wid] = s2; }
    __syncthreads();
    float ts = 0.0f, ts2 = 0.0f;
    const int nw = blockDim.x >> 5;
    for (int w = 0; w < nw; ++w) { ts += sh0[w]; ts2 += sh1[w]; }
    const float mean = ts / (float)N;
    const float var  = ts2 / (float)N - mean * mean;
    const float rstd = rsqrtf(var + 1e-5f);
    for (int i = threadIdx.x; i < N; i += blockDim.x) {
        float v = (r[i] - mean) * rstd * g[i] + bt[i];
        out[(size_t)row * N + i] = gelu_f(v);
    }
}

// ---------------------------------------------------------------------------
// Ripple decode. grid = (B, SEQ_CHUNKS). One wave32 per sequence position:
// lane owns state indices {lane, lane+32}; butterfly shuffles reduce the
// 64-wide sin-contraction. Per-batch params (24897 f32 = 99.6KB) staged to
// LDS by the Tensor Data Mover (wave 0 issues, all waves sync after).
// ---------------------------------------------------------------------------
#define SEQ_CHUNKS 64

__global__ __launch_bounds__(256)
void ripple_kernel(const float* __restrict__ mlp_out,   // [4, MLP_OUT_C]
                   const float* __restrict__ lc_w,      // [RH]
                   const float* __restrict__ lc_b,      // [RH]
                   const float* __restrict__ byp_w,     // [2]
                   const float* __restrict__ byp_b,     // [2]
                   float* __restrict__ out)             // [4, SEQ]
{
    extern __shared__ float sp[];                        // MLP_OUT_C floats
    const int b     = blockIdx.x;
    const int chunk = blockIdx.y;
    const float* P  = mlp_out + (size_t)b * MLP_OUT_C;

    // --- TDM stage: one async tensor DMA, global -> LDS offset 0 ---
    if ((threadIdx.x >> 5) == 0) {
        tdm_load_row_to_lds(P, 0u, (unsigned)MLP_OUT_C);
        __builtin_amdgcn_s_wait_tensorcnt(0);            // TENSORcnt == 0
    }
    __syncthreads();

    const int lane = threadIdx.x & 31;
    const int wid  = threadIdx.x >> 5;                   // 0..7
    const int wavesPerBlock = blockDim.x >> 5;
    const int posPerChunk   = SEQ / SEQ_CHUNKS;          // 64

    const float lcw_lo = lc_w[lane],      lcw_hi = lc_w[lane + 32];
    const float lcb_lo = lc_b[lane],      lcb_hi = lc_b[lane + 32];
    const float bw0 = byp_w[0], bw1 = byp_w[1];
    const float bb0 = byp_b[0], bb1 = byp_b[1];

    const int sBeg = chunk * posPerChunk;
    for (int s = sBeg + wid; s < sBeg + posPerChunk; s += wavesPerBlock) {
        const float t = (float)s;                        // MULT == 1
        float x_lo = t * lcw_lo + lcb_lo;                // lc_x[s, lane]
        float x_hi = t * lcw_hi + lcb_hi;                // lc_x[s, lane+32]

        // bypass term from the initial lc_x
        float bp = sinf(bw0 * x_lo + bw1) + sinf(bw0 * x_hi + bw1);
        bp = wave_sum32(bp);
        const float bypv = bb0 + bp * bb1;

        // RL middle ripple layers
        #pragma unroll
        for (int l = 0; l < RL; ++l) {
            const float*  wbase = sp + l * MID_DIM;              // (w0,w1) pairs
            const float*  bbase = wbase + 2 * RH * RH;           // [o*(RH+1)]
            const float2* wp    = (const float2*)wbase;
            float nlo = 0.0f, nhi = 0.0f;
            for (int o = 0; o < RH; ++o) {
                const float2 wl = wp[o * RH + lane];
                const float2 wh = wp[o * RH + lane + 32];
                float part = sinf(wl.x * x_lo + wl.y) * bbase[o * (RH + 1) + 1 + lane]
                           + sinf(wh.x * x_hi + wh.y) * bbase[o * (RH + 1) + 1 + lane + 32];
                part = wave_sum32(part);
                const float val = gelu_f(bbase[o * (RH + 1)] + part);
                // scatter new state: state index o lives on lane (o & 31)
                if (o < 32) { if (lane == o)      nlo = val; }
                else        { if (lane == o - 32) nhi = val; }
            }
            x_lo = nlo; x_hi = nhi;
        }

        // output ripple layer (single output)
        const float*  obase = sp + RL * MID_DIM;
        const float2* owp   = (const float2*)obase;
        const float*  bo    = obase + 2 * RH;
        const float2  wl = owp[lane];
        const float2  wh = owp[lane + 32];
        float part = sinf(wl.x * x_lo + wl.y) * bo[1 + lane]
                   + sinf(wh.x * x_hi + wh.y) * bo[1 + lane + 32];
        part = wave_sum32(part);

        if (lane == 0)
            out[(size_t)b * SEQ + s] = bo[0] + part + bypv;
    }
}

// ---------------------------------------------------------------------------
// Host launcher
// ---------------------------------------------------------------------------
extern "C" void kernel_launch(void* const* d_in, const int* in_sizes, int n_in,
                              void* d_out, int out_size, void* d_ws, size_t ws_size,
                              hipStream_t stream) {
    const float* x    = (const float*)d_in[0];
    const float* w0   = (const float*)d_in[1];
    const float* b0   = (const float*)d_in[2];
    const float* w1   = (const float*)d_in[3];
    const float* b1   = (const float*)d_in[4];
    const float* g1   = (const float*)d_in[5];
    const float* lb1  = (const float*)d_in[6];
    const float* w2   = (const float*)d_in[7];
    const float* b2   = (const float*)d_in[8];
    const float* g2   = (const float*)d_in[9];
    const float* lb2  = (const float*)d_in[10];
    const float* w3   = (const float*)d_in[11];
    const float* b3   = (const float*)d_in[12];
    const float* lcw  = (const float*)d_in[13];
    const float* lcb  = (const float*)d_in[14];
    const float* bypw = (const float*)d_in[15];
    const float* bypb = (const float*)d_in[16];
    float* out = (float*)d_out;

    float* ws = (float*)d_ws;
    float* h0 = ws;                  // [4,1024]
    float* tm = ws + 4 * HID;        // [4,1024] scratch (pre-LN)
    float* h1 = ws + 8 * HID;        // [4,1024]
    float* h2 = ws + 12 * HID;       // [4,1024]
    float* mo = ws + 16 * HID;       // [4, 24897]

    const int WPB = 4;               // waves per block (blockDim 128)
    auto gemmGrid = [](int N, int wpb) { return ((N + 15) / 16 + wpb - 1) / wpb; };

    // Layer 0: h0 = gelu(x @ w0 + b0)
    gemm4_wmma<<<gemmGrid(HID, WPB), 128, 0, stream>>>(x, w0, b0, h0, IN_DIM, HID, 1);
    // Layer 1: h1 = gelu(LN(h0 @ w1 + b1))
    gemm4_wmma<<<gemmGrid(HID, WPB), 128, 0, stream>>>(h0, w1, b1, tm, HID, HID, 0);
    ln_gelu_kernel<<<B_ROWS, 256, 0, stream>>>(tm, h1, g1, lb1, HID);
    // Layer 2: h2 = gelu(LN(h1 @ w2 + b2))
    gemm4_wmma<<<gemmGrid(HID, WPB), 128, 0, stream>>>(h1, w2, b2, tm, HID, HID, 0);
    ln_gelu_kernel<<<B_ROWS, 256, 0, stream>>>(tm, h2, g2, lb2, HID);
    // Layer 3: mo = h2 @ w3 + b3   (streams 100MB of w3 -> bandwidth bound)
    gemm4_wmma<<<gemmGrid(MLP_OUT_C, WPB), 128, 0, stream>>>(h2, w3, b3, mo, HID, MLP_OUT_C, 0);

    // Ripple decode: grid (B, 64), 256 threads, 99.6KB dynamic LDS / block
    ripple_kernel<<<dim3(B_ROWS, SEQ_CHUNKS), 256, MLP_OUT_C * sizeof(float), stream>>>(
        mo, lcw, lcb, bypw, bypb, out);
}